// B2BConv1d_14645838479478
// MI455X (gfx1250) — compile-verified
//
#include <hip/hip_runtime.h>
#include <stdint.h>

// ---------------- problem constants (from reference) ----------------
#define D_MODEL   4096
#define LSEQ      2048
#define KP        3          // proj conv taps
#define KS        7          // short conv taps
#define TPB       256
#define PER_THREAD 8         // outputs per thread (256*8 = 2048 = L)

#define H1        4          // x1 halo storage (needs 2, padded to 4 for 16B alignment)
#define H2        8          // x2/v halo (6 for conv7 + 2 for conv3)

#if defined(__has_builtin)
#  if __has_builtin(__builtin_amdgcn_s_wait_asynccnt)
#    define HAS_WAIT_ASYNC 1
#  endif
#endif

// Async global->LDS copy (CDNA5 GLOBAL_LOAD_ASYNC_TO_LDS_B128, tracked by ASYNCcnt).
// GV mode: vdst = LDS byte address (low 32 bits of flat shared-aperture address),
//          vaddr = 64-bit global address VGPR pair, saddr = off.
__device__ __forceinline__ void async_copy_f4(const float* gsrc, float* ldst) {
  uint32_t lds_off = (uint32_t)(uintptr_t)ldst;   // aperture bits live in [63:32]
  uint64_t gaddr   = (uint64_t)(uintptr_t)gsrc;
  asm volatile("global_load_async_to_lds_b128 %0, %1, off"
               :
               : "v"(lds_off), "v"(gaddr)
               : "memory");
}

__device__ __forceinline__ void wait_async_copies() {
#if defined(HAS_WAIT_ASYNC)
  __builtin_amdgcn_s_wait_asynccnt(0);
#else
  asm volatile("s_wait_asynccnt 0" ::: "memory");
#endif
}

// out[b,d,l] = conv3(x[3d])[l] * sum_j ws[d/16][j] * ( conv3(x[3d+1]) * conv3(x[3d+2]) )[l-6+j]
__global__ __launch_bounds__(TPB) void fused_hyena_b2b(
    const float* __restrict__ x,       // [B, 3D, L]
    const float* __restrict__ w_proj,  // [3D, 1, 3]
    const float* __restrict__ w_short, // [256, 1, 7]
    float* __restrict__ out)           // [B, D, L]
{
  __shared__ float sX1[H1 + LSEQ];     // x1 row, sX1[i] = x1[i - H1]
  __shared__ float sX2[H2 + LSEQ];     // x2 row, sX2[i] = x2[i - H2]
  __shared__ float sV [H2 + LSEQ];     // v  row

  const int tid = threadIdx.x;
  const int bd  = blockIdx.x;          // b*D + d
  const int b   = bd >> 12;            // D = 4096
  const int d   = bd & (D_MODEL - 1);

  const long rowBase = (long)(b * 3 * D_MODEL + 3 * d) * LSEQ;
  const float* gx1 = x + rowBase;
  const float* gx2 = gx1 + LSEQ;
  const float* gv  = gx2 + LSEQ;

  // zero causal halos (disjoint from async-load destinations)
  if (tid < H2) {
    sX2[tid] = 0.0f;
    sV [tid] = 0.0f;
    if (tid < H1) sX1[tid] = 0.0f;
  }

  // stage 3 rows: 2048 floats = 512 float4 per row -> 2 iterations of 256 lanes
  #pragma unroll
  for (int it = 0; it < 2; ++it) {
    const int e = (it * TPB + tid) * 4;          // element offset, 16B aligned
    async_copy_f4(gx1 + e, &sX1[H1 + e]);
    async_copy_f4(gx2 + e, &sX2[H2 + e]);
    async_copy_f4(gv  + e, &sV [H2 + e]);
  }

  // block-uniform filter taps (scalarized by compiler)
  float wp1[KP], wp2[KP], wpv[KP], ws[KS];
  #pragma unroll
  for (int i = 0; i < KP; ++i) {
    wp1[i] = w_proj[(3 * d + 0) * KP + i];
    wp2[i] = w_proj[(3 * d + 1) * KP + i];
    wpv[i] = w_proj[(3 * d + 2) * KP + i];
  }
  #pragma unroll
  for (int j = 0; j < KS; ++j)
    ws[j] = w_short[(d >> 4) * KS + j];          // D/G_SHORT = 16

  wait_async_copies();
  __syncthreads();

  const int l0 = tid * PER_THREAD;               // outputs l0 .. l0+7

  // sliding windows from LDS
  float X2w[PER_THREAD + 8], Vw[PER_THREAD + 8];
  #pragma unroll
  for (int k = 0; k < PER_THREAD + 8; ++k) {
    X2w[k] = sX2[l0 + k];                        // = x2[l0 - 8 + k]
    Vw [k] = sV [l0 + k];
  }

  // z[k] = pregated + conv3 value at seq pos l0-6+k, k = 0..13
  float z[PER_THREAD + KS - 1];
  #pragma unroll
  for (int k = 0; k < PER_THREAD + KS - 1; ++k) {
    float f2 = fmaf(wp2[2], X2w[k + 2], fmaf(wp2[1], X2w[k + 1], wp2[0] * X2w[k]));
    float fv = fmaf(wpv[2], Vw [k + 2], fmaf(wpv[1], Vw [k + 1], wpv[0] * Vw [k]));
    z[k] = f2 * fv;
  }

  float X1w[PER_THREAD + 2];
  #pragma unroll
  for (int m = 0; m < PER_THREAD + 2; ++m)
    X1w[m] = sX1[l0 + 2 + m];                    // = x1[l0 - 2 + m]

  float res[PER_THREAD];
  #pragma unroll
  for (int m = 0; m < PER_THREAD; ++m) {
    float acc = ws[0] * z[m];
    #pragma unroll
    for (int j = 1; j < KS; ++j) acc = fmaf(ws[j], z[m + j], acc);
    float f1 = fmaf(wp1[2], X1w[m + 2], fmaf(wp1[1], X1w[m + 1], wp1[0] * X1w[m]));
    res[m] = f1 * acc;
  }

  float* orow = out + (long)(b * D_MODEL + d) * LSEQ + l0;
  *(float4*)(orow + 0) = make_float4(res[0], res[1], res[2], res[3]);
  *(float4*)(orow + 4) = make_float4(res[4], res[5], res[6], res[7]);
}

extern "C" void kernel_launch(void* const* d_in, const int* in_sizes, int n_in,
                              void* d_out, int out_size, void* d_ws, size_t ws_size,
                              hipStream_t stream) {
  const float* x       = (const float*)d_in[0];   // [B, 3D, L] f32
  const float* w_proj  = (const float*)d_in[1];   // [3D, 1, 3] f32
  const float* w_short = (const float*)d_in[2];   // [256, 1, 7] f32
  float* out           = (float*)d_out;           // [B, D, L] f32

  const int B = in_sizes[0] / (3 * D_MODEL * LSEQ);   // = 2
  dim3 grid(B * D_MODEL);                              // one block per (b, d) row
  fused_hyena_b2b<<<grid, TPB, 0, stream>>>(x, w_proj, w_short, out);
}